// GAT_PN_12541304504495
// MI455X (gfx1250) — compile-verified
//
#include <hip/hip_runtime.h>
#include <hip/hip_bf16.h>

typedef float v2f __attribute__((ext_vector_type(2)));
typedef float v8f __attribute__((ext_vector_type(8)));

#define TILE_M 128
#define TILE_N 64
#define KT 16

enum { EPI_BIAS = 0, EPI_ACC = 1, EPI_COMBINE = 2 };

// Double-buffered LDS tiles. A: [128][16] rows-of-K, B stored transposed as
// [64][16] rows-of-K. Both padded to 17 so 16 consecutive rows hit distinct
// banks (17 coprime with 64) and each WMMA fragment is a contiguous K-pair.
struct SmemT {
  float As[2][TILE_M][KT + 1];
  float Bs[2][TILE_N][KT + 1];
};

// ---------------------------------------------------------------------------
// Global -> register tile fetch (overlaps with compute in the pipeline).
// BT=true means B is stored (N,K) row-major (multiply by W^T).
// ---------------------------------------------------------------------------
template<bool BT>
__device__ inline void load_tile_regs(const float* __restrict__ A, int lda,
                                      const float* __restrict__ B, int ldb,
                                      int kb, int m0, int n0, int tid,
                                      float4& a0, float4& a1, float4& bv)
{
  const int rowA = tid >> 2;
  const int k4   = (tid & 3) * 4;
  a0 = *(const float4*)(A + (size_t)(m0 + rowA) * lda + kb + k4);
  a1 = *(const float4*)(A + (size_t)(m0 + rowA + 64) * lda + kb + k4);
  if constexpr (!BT) {
    const int krow = tid >> 4;
    const int n4   = (tid & 15) * 4;
    bv = *(const float4*)(B + (size_t)(kb + krow) * ldb + n0 + n4);
  } else {
    const int n = tid >> 2;
    bv = *(const float4*)(B + (size_t)(n0 + n) * ldb + kb + k4);
  }
}

template<bool BT>
__device__ inline void store_tile_lds(SmemT& sm, int buf, int tid,
                                      const float4& a0, const float4& a1,
                                      const float4& bv)
{
  const int rowA = tid >> 2;
  const int k4   = (tid & 3) * 4;
  float* pa0 = &sm.As[buf][rowA][k4];
  pa0[0] = a0.x; pa0[1] = a0.y; pa0[2] = a0.z; pa0[3] = a0.w;
  float* pa1 = &sm.As[buf][rowA + 64][k4];
  pa1[0] = a1.x; pa1[1] = a1.y; pa1[2] = a1.z; pa1[3] = a1.w;
  if constexpr (!BT) {
    const int krow = tid >> 4;
    const int n4   = (tid & 15) * 4;
    sm.Bs[buf][n4 + 0][krow] = bv.x;
    sm.Bs[buf][n4 + 1][krow] = bv.y;
    sm.Bs[buf][n4 + 2][krow] = bv.z;
    sm.Bs[buf][n4 + 3][krow] = bv.w;
  } else {
    const int n = tid >> 2;
    float* pb = &sm.Bs[buf][n][k4];
    pb[0] = bv.x; pb[1] = bv.y; pb[2] = bv.z; pb[3] = bv.w;
  }
}

// ---------------------------------------------------------------------------
// One K=16 chunk: fetch ALL fragments first (batches the dscnt waits), then
// fire 16 v_wmma_f32_16x16x4_f32 back-to-back on the 2x2 wave sub-tiles.
// ---------------------------------------------------------------------------
__device__ inline void compute_chunk(const SmemT& sm, int buf,
                                     int wm, int wn, int l16, int lhalf,
                                     v8f acc[2][2])
{
  v2f a[4][2], b[4][2];
  #pragma unroll
  for (int s = 0; s < 4; ++s) {
    const int kc = s * 4 + 2 * lhalf;   // lanes 0-15: K pair (k,k+1); 16-31: (k+2,k+3)
    #pragma unroll
    for (int i = 0; i < 2; ++i) {
      const float* p = &sm.As[buf][wm * 32 + i * 16 + l16][kc];
      a[s][i].x = p[0]; a[s][i].y = p[1];
    }
    #pragma unroll
    for (int j = 0; j < 2; ++j) {
      const float* p = &sm.Bs[buf][wn * 32 + j * 16 + l16][kc];
      b[s][j].x = p[0]; b[s][j].y = p[1];
    }
  }
  #pragma unroll
  for (int s = 0; s < 4; ++s)
    #pragma unroll
    for (int i = 0; i < 2; ++i)
      #pragma unroll
      for (int j = 0; j < 2; ++j)
        acc[i][j] = __builtin_amdgcn_wmma_f32_16x16x4_f32(
            false, a[s][i], false, b[s][j], (short)0, acc[i][j], false, false);
}

// ---------------------------------------------------------------------------
// Generic fp32 WMMA GEMM:  C = epi( A1@B1 [+ A2@B2] + bias1 + bias2 )
// M % 128 == 0, N % 64 == 0, K % 16 == 0 (guaranteed by caller).
// Software-pipelined: global fetch of chunk c+1 overlaps WMMA of chunk c.
// ---------------------------------------------------------------------------
template<int EPI, bool BT1, bool BT2>
__global__ __launch_bounds__(256)
void gemm_f32_wmma(const float* __restrict__ A1, int lda1,
                   const float* __restrict__ B1, int ldb1, int K1,
                   const float* __restrict__ A2, int lda2,
                   const float* __restrict__ B2, int ldb2, int K2,
                   float* __restrict__ C, int ldc,
                   const float* __restrict__ bias1,
                   const float* __restrict__ bias2,
                   const float* __restrict__ xprev,
                   const float* __restrict__ out0,
                   const float* __restrict__ gat_b,
                   const float* __restrict__ rscale,
                   int Nfirst)
{
  __shared__ SmemT sm;

  const int tid   = threadIdx.x;
  const int wave  = tid >> 5;
  const int lane  = tid & 31;
  const int wm    = wave >> 1;   // 0..3
  const int wn    = wave & 1;    // 0..1
  const int l16   = lane & 15;
  const int lhalf = lane >> 4;
  const int m0    = blockIdx.y * TILE_M;
  const int n0    = blockIdx.x * TILE_N;

  v8f acc[2][2];
  #pragma unroll
  for (int i = 0; i < 2; ++i)
    #pragma unroll
    for (int j = 0; j < 2; ++j)
      #pragma unroll
      for (int r = 0; r < 8; ++r)
        acc[i][j][r] = 0.0f;

  const int nk1 = K1 / KT;
  const int nk2 = A2 ? (K2 / KT) : 0;
  const int nk  = nk1 + nk2;

  float4 ra0, ra1, rb;
  load_tile_regs<BT1>(A1, lda1, B1, ldb1, 0, m0, n0, tid, ra0, ra1, rb);
  store_tile_lds<BT1>(sm, 0, tid, ra0, ra1, rb);
  __syncthreads();

  for (int c = 0; c < nk; ++c) {
    const int  buf     = c & 1;
    const bool haveNxt = (c + 1 < nk);
    const bool nxtIs1  = (c + 1 < nk1);
    if (haveNxt) {
      if (nxtIs1) load_tile_regs<BT1>(A1, lda1, B1, ldb1, (c + 1) * KT,       m0, n0, tid, ra0, ra1, rb);
      else        load_tile_regs<BT2>(A2, lda2, B2, ldb2, (c + 1 - nk1) * KT, m0, n0, tid, ra0, ra1, rb);
    }
    compute_chunk(sm, buf, wm, wn, l16, lhalf, acc);
    if (haveNxt) {
      if (nxtIs1) store_tile_lds<BT1>(sm, buf ^ 1, tid, ra0, ra1, rb);
      else        store_tile_lds<BT2>(sm, buf ^ 1, tid, ra0, ra1, rb);
    }
    __syncthreads();
  }

  float r_ = 0.0f;
  if constexpr (EPI == EPI_COMBINE) r_ = rscale[0];

  #pragma unroll
  for (int i = 0; i < 2; ++i) {
    #pragma unroll
    for (int j = 0; j < 2; ++j) {
      int n = n0 + wn * 32 + j * 16 + l16;
      float b1v = bias1 ? bias1[n] : 0.0f;
      float b2v = bias2 ? bias2[n] : 0.0f;
      #pragma unroll
      for (int r = 0; r < 8; ++r) {
        int m = m0 + wm * 32 + i * 16 + lhalf * 8 + r;
        float val = acc[i][j][r] + b1v + b2v;
        size_t idx = (size_t)m * ldc + n;
        if constexpr (EPI == EPI_BIAS) {
          C[idx] = val;
        } else if constexpr (EPI == EPI_ACC) {
          C[idx] += val;
        } else {
          // x_next = r*(A@W+b) + (1-r)*relu(gat_out) + xprev (in-place over gat proj)
          float rel_in = ((m < Nfirst) ? out0[(size_t)m * ldc + n] : C[idx]) + gat_b[n];
          float rel = rel_in > 0.0f ? rel_in : 0.0f;
          C[idx] = r_ * val + (1.0f - r_) * rel + xprev[idx];
        }
      }
    }
  }
}

// ---------------------------------------------------------------------------
// Small helper kernels
// ---------------------------------------------------------------------------
__device__ inline float wred_sum(float v) {
  #pragma unroll
  for (int o = 16; o > 0; o >>= 1) v += __shfl_xor(v, o, 32);
  return v;
}
__device__ inline float sigf(float x) { return 1.0f / (1.0f + __expf(-x)); }

// out[m][n] = in[m][0]*W[0][n] + in[m][1]*W[1][n] + b[n]   (DIN == 2, D == 256)
__global__ void embed2_kernel(const float* __restrict__ in, const float* __restrict__ W,
                              const float* __restrict__ b, float* __restrict__ out) {
  int idx = blockIdx.x * blockDim.x + threadIdx.x;
  int m = idx >> 8, n = idx & 255;
  out[idx] = in[m * 2] * W[n] + in[m * 2 + 1] * W[256 + n] + b[n];
}

__global__ void init_hc_kernel(const float* __restrict__ h0, const float* __restrict__ c0,
                               float* __restrict__ h, float* __restrict__ c) {
  int idx = blockIdx.x * blockDim.x + threadIdx.x;
  h[idx] = h0[idx & 255];
  c[idx] = c0[idx & 255];
}

// g: (512,1024) -> update h,c (512,256)
__global__ void lstm_gate_kernel(const float* __restrict__ g,
                                 float* __restrict__ h, float* __restrict__ c) {
  int idx = blockIdx.x * blockDim.x + threadIdx.x;
  int b = idx >> 8, d = idx & 255;
  const float* gr = g + (size_t)b * 1024;
  float i_ = sigf(gr[d]);
  float f_ = sigf(gr[256 + d]);
  float gg = tanhf(gr[512 + d]);
  float o_ = sigf(gr[768 + d]);
  float cn = f_ * c[idx] + i_ * gg;
  c[idx] = cn;
  h[idx] = o_ * tanhf(cn);
}

// a_s[j,h] = x0[j,h,:].asrc[h,:]; a_d likewise. grid (200,2), block 32.
__global__ void attn_score_kernel(const float* __restrict__ xw,
                                  const float* __restrict__ asrc,
                                  const float* __restrict__ adst,
                                  float* __restrict__ a_s, float* __restrict__ a_d) {
  int j = blockIdx.x, h = blockIdx.y, lane = threadIdx.x;
  const float* x0 = xw + (size_t)j * 256 + h * 128;
  float s1 = 0.0f, s2 = 0.0f;
  #pragma unroll
  for (int i = 0; i < 4; ++i) {
    int d = lane + 32 * i;
    float xv = x0[d];
    s1 += xv * asrc[h * 128 + d];
    s2 += xv * adst[h * 128 + d];
  }
  s1 = wred_sum(s1); s2 = wred_sum(s2);
  if (lane == 0) { a_s[j * 2 + h] = s1; a_d[j * 2 + h] = s2; }
}

// out0[i,h,d] = softmax_j(leaky(a_s[j,h]+a_d[i,h])) . x0[j,h,d]. grid 200, block 128.
__global__ void attn_aggregate_kernel(const float* __restrict__ xw,
                                      const float* __restrict__ a_s,
                                      const float* __restrict__ a_d,
                                      float* __restrict__ out0) {
  int i = blockIdx.x, t = threadIdx.x;
  #pragma unroll
  for (int h = 0; h < 2; ++h) {
    float ad = a_d[i * 2 + h];
    float mx = -3.0e38f;
    for (int j = 0; j < 200; ++j) {
      float e = a_s[j * 2 + h] + ad;
      e = e > 0.0f ? e : 0.2f * e;
      mx = fmaxf(mx, e);
    }
    float ssum = 0.0f, acc = 0.0f;
    for (int j = 0; j < 200; ++j) {
      float e = a_s[j * 2 + h] + ad;
      e = e > 0.0f ? e : 0.2f * e;
      float w = __expf(e - mx);
      ssum += w;
      acc += w * xw[(size_t)j * 256 + h * 128 + t];
    }
    out0[(size_t)i * 256 + h * 128 + t] = acc / ssum;
  }
}

// i=sig(s0), f=sig(s1), c_new = f*c + i*tanh(s2)
__global__ void enc_cell1_kernel(const float* __restrict__ s0, const float* __restrict__ s1,
                                 const float* __restrict__ s2, const float* __restrict__ c,
                                 float* __restrict__ c_new) {
  int idx = blockIdx.x * blockDim.x + threadIdx.x;
  float i_ = sigf(s0[idx]);
  float f_ = sigf(s1[idx]);
  c_new[idx] = f_ * c[idx] + i_ * tanhf(s2[idx]);
}

// h_new = sig(s3) * tanh(c_new)
__global__ void enc_cell2_kernel(const float* __restrict__ s3, const float* __restrict__ c_new,
                                 float* __restrict__ h_new) {
  int idx = blockIdx.x * blockDim.x + threadIdx.x;
  h_new[idx] = sigf(s3[idx]) * tanhf(c_new[idx]);
}

// u[b,n] = sum_d tanh(q[b,d] + ref[b*N+n,d]) * v[d]; one wave per (b,n)
__global__ void u_kernel(const float* __restrict__ q, const float* __restrict__ ref,
                         const float* __restrict__ v, const float* __restrict__ mask,
                         float* __restrict__ lat, float* __restrict__ um) {
  int w = blockIdx.x * 8 + (threadIdx.x >> 5);
  int lane = threadIdx.x & 31;
  int b = w / 200;
  const float* qr = q + (size_t)b * 256;
  const float* rr = ref + (size_t)w * 256;
  float s = 0.0f;
  #pragma unroll
  for (int i = 0; i < 8; ++i) {
    int d = lane + 32 * i;
    s += tanhf(qr[d] + rr[d]) * v[d];
  }
  s = wred_sum(s);
  if (lane == 0) {
    lat[w] = s;
    um[w] = 10.0f * tanhf(s) + mask[w];
  }
}

// row softmax over N=200; one wave per row, 512 rows
__global__ void softmax_kernel(const float* __restrict__ u, float* __restrict__ out) {
  int row = blockIdx.x * 8 + (threadIdx.x >> 5);
  int lane = threadIdx.x & 31;
  const float* ur = u + (size_t)row * 200;
  float vals[7];
  float mx = -3.0e38f;
  #pragma unroll
  for (int i = 0; i < 7; ++i) {
    int n = lane + 32 * i;
    float vv = (n < 200) ? ur[n] : -3.0e38f;
    vals[i] = vv;
    mx = fmaxf(mx, vv);
  }
  #pragma unroll
  for (int o = 16; o > 0; o >>= 1) mx = fmaxf(mx, __shfl_xor(mx, o, 32));
  float s = 0.0f;
  #pragma unroll
  for (int i = 0; i < 7; ++i) {
    int n = lane + 32 * i;
    if (n < 200) { vals[i] = __expf(vals[i] - mx); s += vals[i]; }
  }
  s = wred_sum(s);
  float inv = 1.0f / s;
  #pragma unroll
  for (int i = 0; i < 7; ++i) {
    int n = lane + 32 * i;
    if (n < 200) out[(size_t)row * 200 + n] = vals[i] * inv;
  }
}

// ---------------------------------------------------------------------------
extern "C" void kernel_launch(void* const* d_in, const int* in_sizes, int n_in,
                              void* d_out, int out_size, void* d_ws, size_t ws_size,
                              hipStream_t stream) {
  const int B = 512, N = 200, D = 256;
  const int BN = B * N;            // 102400
  const int BND = BN * D;          // 26,214,400

  const float* x        = (const float*)d_in[0];
  const float* X_all    = (const float*)d_in[1];
  const float* mask     = (const float*)d_in[2];
  const float* emb_x_W  = (const float*)d_in[3];
  const float* emb_x_b  = (const float*)d_in[4];
  const float* emb_a_W  = (const float*)d_in[5];
  const float* emb_a_b  = (const float*)d_in[6];
  const float* h0       = (const float*)d_in[7];
  const float* c0       = (const float*)d_in[8];
  const float* Wih      = (const float*)d_in[9];
  const float* Whh      = (const float*)d_in[10];
  const float* bih      = (const float*)d_in[11];
  const float* bhh      = (const float*)d_in[12];
  const float* gat1_W   = (const float*)d_in[13];
  const float* gat1_as  = (const float*)d_in[14];
  const float* gat1_ad  = (const float*)d_in[15];
  const float* gat1_b   = (const float*)d_in[16];
  const float* gat2_W   = (const float*)d_in[17];
  const float* gat2_as  = (const float*)d_in[18];
  const float* gat2_ad  = (const float*)d_in[19];
  const float* gat2_b   = (const float*)d_in[20];
  const float* r1       = (const float*)d_in[21];
  const float* r2       = (const float*)d_in[22];
  const float* W1       = (const float*)d_in[23];
  const float* b1       = (const float*)d_in[24];
  const float* W2       = (const float*)d_in[25];
  const float* b2       = (const float*)d_in[26];
  const float* enc_Wx   = (const float*)d_in[27];
  const float* enc_bx   = (const float*)d_in[28];
  const float* enc_Wh   = (const float*)d_in[29];
  const float* enc_bh   = (const float*)d_in[30];
  const float* enc_Wc   = (const float*)d_in[31];
  const float* enc_bc   = (const float*)d_in[32];
  const float* v        = (const float*)d_in[33];
  const float* Wref_W   = (const float*)d_in[34];
  const float* Wref_b   = (const float*)d_in[35];
  const float* Wq_W     = (const float*)d_in[36];
  const float* Wq_b     = (const float*)d_in[37];

  float* ws = (float*)d_ws;
  float* BIG0 = ws;                 // ctx(cf) -> xw2 -> x2
  float* BIG1 = ws + (size_t)BND;   // xw1 -> x1 -> ref
  float* xe   = BIG1 + (size_t)BND;
  float* hbuf = xe + B * D;
  float* cbuf = hbuf + B * D;
  float* gbuf = cbuf + B * D;           // 512*1024
  float* out0 = gbuf + B * 4 * D;       // 200*256
  float* a_s  = out0 + N * D;
  float* a_d  = a_s + 512;
  float* sbuf = a_d + 512;              // 4 * 512*256
  float* qbuf = sbuf + 4 * B * D;
  float* um   = qbuf + B * D;           // 512*200

  float* out_soft = (float*)d_out;
  float* out_h    = out_soft + BN;
  float* out_c    = out_h + B * D;
  float* out_lat  = out_c + B * D;

  const float* NUL = nullptr;

  // 1. embeddings
  embed2_kernel<<<(B * D) / 256, 256, 0, stream>>>(x, emb_x_W, emb_x_b, xe);
  embed2_kernel<<<(BN * D) / 256, 256, 0, stream>>>(X_all, emb_a_W, emb_a_b, BIG0);
  init_hc_kernel<<<(B * D) / 256, 256, 0, stream>>>(h0, c0, hbuf, cbuf);

  // 2. LSTM scan: g = xt@Wih^T + h@Whh^T + bih + bhh, then gates
  for (int t = 0; t < N; ++t) {
    gemm_f32_wmma<EPI_BIAS, true, true><<<dim3((4 * D) / TILE_N, B / TILE_M), 256, 0, stream>>>(
        BIG0 + (size_t)t * D, N * D, Wih, D, D,
        hbuf, D, Whh, D, D,
        gbuf, 4 * D, bih, bhh, NUL, NUL, NUL, NUL, 0);
    lstm_gate_kernel<<<(B * D) / 256, 256, 0, stream>>>(gbuf, hbuf, cbuf);
  }

  // 3. GAT layer 1
  gemm_f32_wmma<EPI_BIAS, false, false><<<dim3(D / TILE_N, BN / TILE_M), 256, 0, stream>>>(
      BIG0, D, gat1_W, D, D, NUL, 0, NUL, 0, 0,
      BIG1, D, NUL, NUL, NUL, NUL, NUL, NUL, 0);
  attn_score_kernel<<<dim3(N, 2), 32, 0, stream>>>(BIG1, gat1_as, gat1_ad, a_s, a_d);
  attn_aggregate_kernel<<<N, 128, 0, stream>>>(BIG1, a_s, a_d, out0);
  gemm_f32_wmma<EPI_COMBINE, false, false><<<dim3(D / TILE_N, BN / TILE_M), 256, 0, stream>>>(
      BIG0, D, W1, D, D, NUL, 0, NUL, 0, 0,
      BIG1, D, b1, NUL, /*xprev*/ BIG0, out0, gat1_b, r1, N);   // x1 in BIG1

  // 4. GAT layer 2
  gemm_f32_wmma<EPI_BIAS, false, false><<<dim3(D / TILE_N, BN / TILE_M), 256, 0, stream>>>(
      BIG1, D, gat2_W, D, D, NUL, 0, NUL, 0, 0,
      BIG0, D, NUL, NUL, NUL, NUL, NUL, NUL, 0);
  attn_score_kernel<<<dim3(N, 2), 32, 0, stream>>>(BIG0, gat2_as, gat2_ad, a_s, a_d);
  attn_aggregate_kernel<<<N, 128, 0, stream>>>(BIG0, a_s, a_d, out0);
  gemm_f32_wmma<EPI_COMBINE, false, false><<<dim3(D / TILE_N, BN / TILE_M), 256, 0, stream>>>(
      BIG1, D, W2, D, D, NUL, 0, NUL, 0, 0,
      BIG0, D, b2, NUL, /*xprev*/ BIG1, out0, gat2_b, r2, N);   // x2 in BIG0

  // 5. ref = x2 @ Wref_W + Wref_b  -> BIG1
  gemm_f32_wmma<EPI_BIAS, false, false><<<dim3(D / TILE_N, BN / TILE_M), 256, 0, stream>>>(
      BIG0, D, Wref_W, D, D, NUL, 0, NUL, 0, 0,
      BIG1, D, Wref_b, NUL, NUL, NUL, NUL, NUL, 0);

  // 6. encoder attention-LSTM cell (512x256 GEMMs)
  dim3 gsmall(D / TILE_N, B / TILE_M);
  for (int k = 0; k < 4; ++k)
    gemm_f32_wmma<EPI_BIAS, false, false><<<gsmall, 256, 0, stream>>>(
        xe, D, enc_Wx + (size_t)k * D * D, D, D, NUL, 0, NUL, 0, 0,
        sbuf + (size_t)k * B * D, D, enc_bx + k * D, NUL, NUL, NUL, NUL, NUL, 0);
  for (int k = 0; k < 4; ++k)
    gemm_f32_wmma<EPI_ACC, false, false><<<gsmall, 256, 0, stream>>>(
        hbuf, D, enc_Wh + (size_t)k * D * D, D, D, NUL, 0, NUL, 0, 0,
        sbuf + (size_t)k * B * D, D, enc_bh + k * D, NUL, NUL, NUL, NUL, NUL, 0);
  for (int k = 0; k < 2; ++k)
    gemm_f32_wmma<EPI_ACC, false, false><<<gsmall, 256, 0, stream>>>(
        cbuf, D, enc_Wc + (size_t)k * D * D, D, D, NUL, 0, NUL, 0, 0,
        sbuf + (size_t)k * B * D, D, enc_bc + k * D, NUL, NUL, NUL, NUL, NUL, 0);
  enc_cell1_kernel<<<(B * D) / 256, 256, 0, stream>>>(
      sbuf, sbuf + B * D, sbuf + 2 * B * D, cbuf, out_c);
  gemm_f32_wmma<EPI_ACC, false, false><<<gsmall, 256, 0, stream>>>(
      out_c, D, enc_Wc + (size_t)2 * D * D, D, D, NUL, 0, NUL, 0, 0,
      sbuf + (size_t)3 * B * D, D, enc_bc + 2 * D, NUL, NUL, NUL, NUL, NUL, 0);
  enc_cell2_kernel<<<(B * D) / 256, 256, 0, stream>>>(sbuf + (size_t)3 * B * D, out_c, out_h);

  // 7. q = h_new @ Wq_W + Wq_b
  gemm_f32_wmma<EPI_BIAS, false, false><<<gsmall, 256, 0, stream>>>(
      out_h, D, Wq_W, D, D, NUL, 0, NUL, 0, 0,
      qbuf, D, Wq_b, NUL, NUL, NUL, NUL, NUL, 0);

  // 8. u / latent_u, then softmax
  u_kernel<<<BN / 8, 256, 0, stream>>>(qbuf, BIG1, v, mask, out_lat, um);
  softmax_kernel<<<B / 8, 256, 0, stream>>>(um, out_soft);
}